// SmoothCVaRLoss_48473000902892
// MI455X (gfx1250) — compile-verified
//
#include <hip/hip_runtime.h>
#include <math.h>

// ---------------------------------------------------------------------------
// CVaR dual on MI455X (gfx1250, wave32).
//
// Algorithm (exact reformulation of the reference):
//   losses l_i = -pnl_i,  t_j = tmin + j*dt (j=0..99, dt=(tmax-tmin)/99)
//   sum_i relu(l_i - t_j) = SufS_j - t_j * SufC_j
//     where bin(l) = floor((l - tmin)/dt), S_k = sum of l in bin k,
//     C_k = count in bin k, Suf*_j = sum_{k>=j} *_k.
//   cvar_j = t_j + (SufS_j - t_j*SufC_j) / (0.05*N);  answer = min_j cvar_j
//
// Passes: (1) min/max streaming reduction, (2) 100-bin histogram via LDS
// atomics, (3) suffix sums as an fp32 WMMA matmul against an upper-
// triangular ones matrix (V_WMMA_F32_16X16X4_F32), single wave.
//
// ws layout (32-bit words): [0]=min(ord) [1]=max(ord)
//                           [2..101]=bin counts (u32)
//                           [102..201]=bin sums (f32)
// ---------------------------------------------------------------------------

typedef __attribute__((ext_vector_type(2))) float v2f;
typedef __attribute__((ext_vector_type(8))) float v8f;

#define NS    100          // N_SAMPLES
#define NPAD  112          // padded to multiple of 16 for WMMA tiling
#define INV_1MA 20.0f      // 1/(1-ALPHA) = 1/0.05

// Order-preserving float <-> uint mapping (unsigned compare == float compare)
__device__ __forceinline__ unsigned f2ord(float f) {
    unsigned u = __float_as_uint(f);
    return (u & 0x80000000u) ? ~u : (u | 0x80000000u);
}
__device__ __forceinline__ float ord2f(unsigned u) {
    unsigned b = (u & 0x80000000u) ? (u ^ 0x80000000u) : ~u;
    return __uint_as_float(b);
}

// ---------------------------------------------------------------- init ------
__global__ void cvar_init(unsigned* __restrict__ ws) {
    int t = threadIdx.x;
    if (t == 0) { ws[0] = 0xFFFFFFFFu; ws[1] = 0u; }
    for (int i = t; i < 2 * NS; i += blockDim.x) ws[2 + i] = 0u;  // 0u bits == 0.0f
}

// ------------------------------------------------------------- min/max ------
__global__ void cvar_minmax(const float* __restrict__ pnl, int n,
                            unsigned* __restrict__ ws) {
    float lmin =  INFINITY;
    float lmax = -INFINITY;
    const int stride = gridDim.x * blockDim.x;
    const int tid    = blockIdx.x * blockDim.x + threadIdx.x;

    const int n4 = n >> 2;
    const float4* p4 = (const float4*)pnl;
    for (int k = tid; k < n4; k += stride) {
        float4 v = p4[k];
        float a = -v.x, b = -v.y, c = -v.z, d = -v.w;
        lmin = fminf(lmin, fminf(fminf(a, b), fminf(c, d)));
        lmax = fmaxf(lmax, fmaxf(fmaxf(a, b), fmaxf(c, d)));
    }
    for (int k = (n4 << 2) + tid; k < n; k += stride) {
        float l = -pnl[k];
        lmin = fminf(lmin, l);
        lmax = fmaxf(lmax, l);
    }

    // wave32 butterfly reduce
    for (int m = 16; m > 0; m >>= 1) {
        lmin = fminf(lmin, __shfl_xor(lmin, m, 32));
        lmax = fmaxf(lmax, __shfl_xor(lmax, m, 32));
    }
    __shared__ float smin[8], smax[8];
    const int wid = threadIdx.x >> 5;
    if ((threadIdx.x & 31) == 0) { smin[wid] = lmin; smax[wid] = lmax; }
    __syncthreads();
    if (threadIdx.x == 0) {
        const int nw = blockDim.x >> 5;
        float bmin = smin[0], bmax = smax[0];
        for (int w = 1; w < nw; ++w) {
            bmin = fminf(bmin, smin[w]);
            bmax = fmaxf(bmax, smax[w]);
        }
        atomicMin(&ws[0], f2ord(bmin));
        atomicMax(&ws[1], f2ord(bmax));
    }
}

// ----------------------------------------------------------- histogram ------
__global__ void cvar_hist(const float* __restrict__ pnl, int n,
                          unsigned* __restrict__ ws) {
    __shared__ unsigned scnt[NS];
    __shared__ float    ssum[NS];
    for (int i = threadIdx.x; i < NS; i += blockDim.x) { scnt[i] = 0u; ssum[i] = 0.0f; }
    __syncthreads();

    const float tmin = ord2f(ws[0]);
    const float tmax = ord2f(ws[1]);
    const float dt   = (tmax - tmin) / (float)(NS - 1);
    const float inv_dt = (dt > 0.0f) ? (1.0f / dt) : 0.0f;

    const int stride = gridDim.x * blockDim.x;
    const int tid    = blockIdx.x * blockDim.x + threadIdx.x;

    const int n4 = n >> 2;
    const float4* p4 = (const float4*)pnl;
    for (int k = tid; k < n4; k += stride) {
        float4 v = p4[k];
        float l[4] = { -v.x, -v.y, -v.z, -v.w };
#pragma unroll
        for (int e = 0; e < 4; ++e) {
            int b = (int)((l[e] - tmin) * inv_dt);
            b = min(NS - 1, max(0, b));
            atomicAdd(&scnt[b], 1u);      // ds_add_u32
            atomicAdd(&ssum[b], l[e]);    // ds_add_f32
        }
    }
    for (int k = (n4 << 2) + tid; k < n; k += stride) {
        float l = -pnl[k];
        int b = (int)((l - tmin) * inv_dt);
        b = min(NS - 1, max(0, b));
        atomicAdd(&scnt[b], 1u);
        atomicAdd(&ssum[b], l);
    }
    __syncthreads();

    unsigned* gcnt = ws + 2;
    float*    gsum = (float*)(ws + 2 + NS);
    for (int i = threadIdx.x; i < NS; i += blockDim.x) {
        if (scnt[i]) {
            atomicAdd(&gcnt[i], scnt[i]);
            atomicAdd(&gsum[i], ssum[i]);
        }
    }
}

// ------------------------------------------------- finalize (WMMA f32) ------
// Single wave (32 threads). Suffix sums of S (bin sums) and C (bin counts)
// computed as [S;C](2x112, zero padded) x U(112x112), U[k][j] = (k >= j),
// tiled as V_WMMA_F32_16X16X4_F32: 7 N-tiles x 28 K-steps.
//
// Branch-free fragment construction: every lane loads both S and C words
// (uniform control flow -> plain ds_load, no exec-mask save/restore around
// the WMMA region) and selects its A-row contribution with arithmetic masks.
__global__ void cvar_final(const unsigned* __restrict__ ws, int n,
                           float* __restrict__ out) {
    __shared__ float sS[NPAD];
    __shared__ float sC[NPAD];
    const int lane = threadIdx.x;          // 0..31, one wave
    const float* gsum = (const float*)(ws + 2 + NS);
    for (int i = lane; i < NPAD; i += 32) {
        sS[i] = (i < NS) ? gsum[i] : 0.0f;
        sC[i] = (i < NS) ? (float)ws[2 + i] : 0.0f;
    }
    __syncthreads();

    const float tmin  = ord2f(ws[0]);
    const float tmax  = ord2f(ws[1]);
    const float dt    = (tmax - tmin) / (float)(NS - 1);
    const float invmn = INV_1MA / (float)n;     // 1 / (0.05 * N)

    const int hi  = (lane >= 16) ? 1 : 0;
    const int l16 = lane & 15;
    // Arithmetic row masks: row 0 of A carries S, row 1 carries C, rest zero.
    const float mS = (l16 == 0) ? 1.0f : 0.0f;
    const float mC = (l16 == 1) ? 1.0f : 0.0f;
    const bool  valid = (lane < 16) && (l16 < NS % 16 || true); // refined per tile below
    float best = INFINITY;

    for (int nt = 0; nt < NPAD / 16; ++nt) {
        v8f acc = {};
        const int j = nt * 16 + l16;            // B column / output column
        for (int kt = 0; kt < NPAD / 4; ++kt) {
            const int ka = kt * 4 + 2 * hi;     // K index held by VGPR0 of this lane
            // Unconditional LDS reads (all lanes, uniform flow)
            const float sv0 = sS[ka], sv1 = sS[ka + 1];
            const float cv0 = sC[ka], cv1 = sC[ka + 1];
            // A (16x4 f32): M = l16; row 0 = S, row 1 = C, rows 2..15 = 0
            v2f a;
            a.x = sv0 * mS + cv0 * mC;
            a.y = sv1 * mS + cv1 * mC;
            // B (4x16 f32): U[k][j] = (k >= j)
            v2f b;
            b.x = (ka     >= j) ? 1.0f : 0.0f;
            b.y = (ka + 1 >= j) ? 1.0f : 0.0f;
            acc = __builtin_amdgcn_wmma_f32_16x16x4_f32(
                false, a, false, b, (short)0, acc, false, false);
        }
        // D layout: lanes 0-15, VGPR v -> row M=v.  acc[0]=SufS_j, acc[1]=SufC_j
        // Compute unconditionally, mask with a select (no exec divergence).
        const float sufS = acc[0];
        const float sufC = acc[1];
        const float t    = tmin + dt * (float)j;
        const float cv   = t + (sufS - t * sufC) * invmn;
        const bool  take = (lane < 16) && (j < NS);
        best = take ? fminf(best, cv) : best;
        (void)valid;
    }
    for (int m = 16; m > 0; m >>= 1)
        best = fminf(best, __shfl_xor(best, m, 32));
    if (lane == 0) out[0] = best;
}

// ----------------------------------------------------------------------------
extern "C" void kernel_launch(void* const* d_in, const int* in_sizes, int n_in,
                              void* d_out, int out_size, void* d_ws, size_t ws_size,
                              hipStream_t stream) {
    const float* pnl = (const float*)d_in[0];
    const int n = in_sizes[0];
    unsigned* ws = (unsigned*)d_ws;
    float* out = (float*)d_out;

    const int threads = 256;                         // 8 wave32s per block
    int blocks = (n / 4 + threads - 1) / threads;
    if (blocks > 1024) blocks = 1024;
    if (blocks < 1) blocks = 1;

    cvar_init  <<<1,      256,     0, stream>>>(ws);
    cvar_minmax<<<blocks, threads, 0, stream>>>(pnl, n, ws);
    cvar_hist  <<<blocks, threads, 0, stream>>>(pnl, n, ws);
    cvar_final <<<1,      32,      0, stream>>>(ws, n, out);
}